// InnerProductDecoder2_26414048870609
// MI455X (gfx1250) — compile-verified
//
#include <hip/hip_runtime.h>
#include <hip/hip_bf16.h>
#include <math.h>

// CDNA5 / gfx1250: wave32, WMMA f32 16x16x4 (full-precision f32 matrix op).
typedef __attribute__((ext_vector_type(2))) float v2f;
typedef __attribute__((ext_vector_type(8))) float v8f;

__device__ __forceinline__ unsigned hash_u32(unsigned x) {
    // lowbias32 finalizer-style hash (counter-based RNG)
    x ^= x >> 16; x *= 0x7feb352dU;
    x ^= x >> 15; x *= 0x846ca68bU;
    x ^= x >> 16;
    return x;
}

__device__ __forceinline__ float gumbel_from_u32(unsigned h) {
    float u = (float)(h >> 8) * 5.9604645e-08f;   // uniform in [0,1)
    u = fmaxf(u, 1e-10f);                         // match reference minval=1e-10
    return -__logf(-__logf(u));                   // Gumbel(0,1)
}

// One wave handles 16 edges: dot(z1[e0_i], z1[e1_i]) = diag of A(16x128)*B(128x16),
// computed with 32 chained V_WMMA_F32_16X16X4_F32 ops (full fp32 precision).
__global__ void ipd2_edge_gate_wmma(const float* __restrict__ z1,
                                    const float* __restrict__ z2,
                                    const int*   __restrict__ ei,   // [2,E] flat
                                    const int*   __restrict__ temp, // scalar
                                    float*       __restrict__ out,
                                    int E)
{
    const int lane          = threadIdx.x & 31;
    const int waveInBlock   = threadIdx.x >> 5;
    const int wavesPerBlock = blockDim.x >> 5;
    const long tile = (long)blockIdx.x * wavesPerBlock + waveInBlock;
    const long base = tile * 16;
    if (base >= E) return;   // wave-uniform exit -> EXEC all-1s for WMMA

    // A-matrix f32 16x4 layout: lanes 0-15 and 16-31 both map to M = lane&15;
    // VGPR0/1 hold K = 2*(lane>>4) and +1 within each 4-wide K chunk.
    // B-matrix 4x16 mirrors it: N = lane&15, same K-pair split across lane halves.
    const int m  = lane & 15;
    const int kb = (lane >> 4) << 1;   // 0 or 2

    long e  = base + m;
    long ec = (e < E) ? e : (long)(E - 1);   // tail clamp (E%16==0 in practice)

    const int r0 = ei[ec];              // edge source row
    const int r1 = ei[(long)E + ec];    // edge dest row

    const float* a_row = z1 + (long)r0 * 128 + kb;   // A: rows z1[e0]
    const float* b_row = z1 + (long)r1 * 128 + kb;   // B: cols z1[e1]

    v8f acc = {};
    #pragma unroll 8
    for (int kc = 0; kc < 128; kc += 4) {
        v2f av = *(const v2f*)(a_row + kc);          // global_load_b64
        v2f bv = *(const v2f*)(b_row + kc);          // global_load_b64
        // v_wmma_f32_16x16x4_f32: D = A*B + C, fp32 throughout
        acc = __builtin_amdgcn_wmma_f32_16x16x4_f32(
                  /*neg_a=*/false, av, /*neg_b=*/false, bv,
                  /*c_mod=*/(short)0, acc, /*reuse_a=*/false, /*reuse_b=*/false);
    }

    // Extract diagonal D[m][m] -> lane m.
    // C/D layout: VGPR j holds (M=j, N=lane) for lanes 0-15 and (M=j+8, N=lane-16)
    // for lanes 16-31  =>  (m,m) lives in VGPR (m&7) at lane m (m<8) or m+16 (m>=8).
    const int jv      = m & 7;
    const int srcLane = (m < 8) ? m : (m + 16);
    float vf = 0.f;
    #pragma unroll
    for (int j = 0; j < 8; ++j) {
        float t = __shfl(acc[j], srcLane, 32);       // full-wave shuffles
        if (jv == j) vf = t;
    }

    if (lane < 16 && e < E) {
        const float tau = (float)temp[0];

        // Deterministic per-edge Gumbel pair (counter-based hash RNG).
        unsigned eh = (unsigned)e;
        float g0 = gumbel_from_u32(hash_u32(eh * 2u + 0x9e3779b9u));
        float g1 = gumbel_from_u32(hash_u32(eh * 2u + 1u + 0x85ebca6bu));

        // softmax over {(vf+g0)/tau, (0+g1)/tau}, component 0:
        float y0 = 1.f / (1.f + __expf((g1 - vf - g0) / tau));
        float yh = (y0 >= 0.5f) ? 1.f : 0.f;         // argmax one-hot (ties -> idx 0)
        float a  = (yh - y0) + y0;                   // straight-through forward value

        float vn = z2[(long)r0 * 2] + z2[(long)r1 * 2];   // z2[e0,0] + z2[e1,0]
        float sf = 1.f / (1.f + __expf(-vf));
        float sn = 1.f / (1.f + __expf(-vn));
        out[e] = a * sf + (1.f - a) * sn;
    }
}

extern "C" void kernel_launch(void* const* d_in, const int* in_sizes, int n_in,
                              void* d_out, int out_size, void* d_ws, size_t ws_size,
                              hipStream_t stream) {
    const float* z1   = (const float*)d_in[0];   // [N,128] f32
    const float* z2   = (const float*)d_in[1];   // [N,2]   f32
    const int*   ei   = (const int*)  d_in[2];   // [2,E]   int
    const int*   temp = (const int*)  d_in[3];   // scalar
    float*       out  = (float*)d_out;           // [E]     f32

    const int E = out_size;                      // == in_sizes[2] / 2
    const int wavesPerBlock = 8;                 // 256 threads = 8 wave32
    const int numTiles = (E + 15) / 16;          // 16 edges per wave
    const int blocks   = (numTiles + wavesPerBlock - 1) / wavesPerBlock;

    ipd2_edge_gate_wmma<<<blocks, wavesPerBlock * 32, 0, stream>>>(
        z1, z2, ei, temp, out, E);
}